// PCconv_56281251446859
// MI455X (gfx1250) — compile-verified
//
#include <hip/hip_runtime.h>
#include <math.h>

// ---- CDNA5 WMMA types: V_WMMA_F32_16X16X4_F32 (wave32) ----
typedef __attribute__((ext_vector_type(2))) float v2f;   // A/B fragment: 16x4 / 4x16 f32 = 2 VGPRs
typedef __attribute__((ext_vector_type(8))) float v8f;   // C/D fragment: 16x16 f32 = 8 VGPRs

#define WMMA_F32X4(a, b, c) \
    __builtin_amdgcn_wmma_f32_16x16x4_f32(false, (a), false, (b), (short)0, (c), false, false)

#define HW 1024   // 32*32
#define CC 512

// ---------------------------------------------------------------------------
// Kernel: gaussian weights G[p][q], p=i*32+k, q=x*32+y, normalized per p.
// ---------------------------------------------------------------------------
__global__ void k_gauss(float* __restrict__ G) {
    const int p = blockIdx.x;             // 0..1023
    const int pi = p >> 5, pk = p & 31;
    const int tid = threadIdx.x;
    __shared__ float red[256];
    float vals[4];
    float acc = 0.f;
#pragma unroll
    for (int t = 0; t < 4; ++t) {
        int q = tid + t * 256;
        int qx = q >> 5, qy = q & 31;
        float dx = (float)(qx - pi), dy = (float)(qy - pk);
        float v = __expf(-(dx * dx + dy * dy) * (1.0f / (2.0f * 1.5f * 1.5f)));
        vals[t] = v;
        acc += v;
    }
    red[tid] = acc;
    __syncthreads();
    for (int s = 128; s > 0; s >>= 1) {
        if (tid < s) red[tid] += red[tid + s];
        __syncthreads();
    }
    float inv = 1.0f / red[0];
#pragma unroll
    for (int t = 0; t < 4; ++t) G[p * HW + tid + t * 256] = vals[t] * inv;
}

// ---------------------------------------------------------------------------
// Kernel: per-channel mean over 32x32 (SE global average pool)
// ---------------------------------------------------------------------------
__global__ void k_mean(const float* __restrict__ x, float* __restrict__ s) {
    const int c = blockIdx.x;
    const int tid = threadIdx.x;
    __shared__ float red[256];
    float a = 0.f;
#pragma unroll
    for (int t = 0; t < 4; ++t) a += x[c * HW + tid + t * 256];
    red[tid] = a;
    __syncthreads();
    for (int st = 128; st > 0; st >>= 1) {
        if (tid < st) red[tid] += red[tid + st];
        __syncthreads();
    }
    if (tid == 0) s[c] = red[0] * (1.0f / 1024.0f);
}

// ---------------------------------------------------------------------------
// Kernel: SE FCs: s1 = relu(W1 @ s), s2 = sigmoid(W2 @ s1). One block, 512 thr.
// ---------------------------------------------------------------------------
__global__ void k_se(const float* __restrict__ s, const float* __restrict__ w1,
                     const float* __restrict__ w2, float* __restrict__ s2) {
    __shared__ float sc[512];
    __shared__ float s1[32];
    const int tid = threadIdx.x;
    sc[tid] = s[tid];
    __syncthreads();
    if (tid < 32) {
        float a = 0.f;
        for (int c = 0; c < 512; ++c) a += w1[tid * 512 + c] * sc[c];
        s1[tid] = fmaxf(a, 0.f);
    }
    __syncthreads();
    float a = 0.f;
#pragma unroll
    for (int j = 0; j < 32; ++j) a += w2[tid * 32 + j] * s1[j];
    s2[tid] = 1.0f / (1.0f + __expf(-a));
}

// ---------------------------------------------------------------------------
// Kernel: out32 = x * s2[c];  sig = sigmoid(out32)
// ---------------------------------------------------------------------------
__global__ void k_scale(const float* __restrict__ x, const float* __restrict__ s2,
                        float* __restrict__ out32, float* __restrict__ sig) {
    const int idx = blockIdx.x * 256 + threadIdx.x;  // 524288 total
    const int c = idx >> 10;
    float v = x[idx] * s2[c];
    out32[idx] = v;
    sig[idx] = 1.0f / (1.0f + __expf(-v));
}

// ---------------------------------------------------------------------------
// GEMM1 (WMMA f32): D[p][c] = sum_q G[p][q] * out32[c][q]
// M=1024(p), N=512(c), K=1024(q). Wave computes 64(M) x 64(N): 16 accumulators
// (128 VGPRs) -- __launch_bounds__(256,1) lifts the compiler's default 128-VGPR
// cap so accumulators stay resident (no scratch spills).
// grid = 16 (M blocks of 64), 256 threads = 8 waves (N blocks of 64).
// The D element at flat index f = p*512+c of the (HW,C) result is scattered
// into catT[q][i] (K-major concat buffer for GEMM2): i = f/1024, q = f%1024.
// ---------------------------------------------------------------------------
__global__ void __launch_bounds__(256, 1)
k_gemm1(const float* __restrict__ G, const float* __restrict__ B,
        float* __restrict__ catT) {
    const int lane = threadIdx.x & 31;
    const int wave = threadIdx.x >> 5;
    const int half = lane >> 4;
    const int row = lane & 15;
    const int m0 = blockIdx.x * 64;
    const int n0 = wave * 64;

    const float* abase = G + (m0 + row) * HW + 2 * half;   // + tm*16*HW per M tile
    const float* bbase = B + (n0 + row) * HW + 2 * half;   // + tn*16*HW per N tile

    v8f acc[4][4];
#pragma unroll
    for (int tm = 0; tm < 4; ++tm)
#pragma unroll
        for (int tn = 0; tn < 4; ++tn) acc[tm][tn] = (v8f){};

    for (int k = 0; k < HW; k += 4) {
        v2f a[4], b[4];
#pragma unroll
        for (int t = 0; t < 4; ++t) {
            a[t] = *(const v2f*)(abase + t * 16 * HW + k);
            b[t] = *(const v2f*)(bbase + t * 16 * HW + k);
        }
#pragma unroll
        for (int tm = 0; tm < 4; ++tm)
#pragma unroll
            for (int tn = 0; tn < 4; ++tn)
                acc[tm][tn] = WMMA_F32X4(a[tm], b[tn], acc[tm][tn]);
    }
    // C/D layout: element (M = 8*half + r, N = row) in VGPR r
#pragma unroll
    for (int tm = 0; tm < 4; ++tm) {
#pragma unroll
        for (int tn = 0; tn < 4; ++tn) {
#pragma unroll
            for (int r = 0; r < 8; ++r) {
                int m = m0 + tm * 16 + 8 * half + r;      // p
                int n = n0 + tn * 16 + row;               // c
                int f = m * CC + n;                       // flat (p,c) index
                catT[((f & 1023) << 10) + (f >> 10)] = acc[tm][tn][r];
            }
        }
    }
}

// ---------------------------------------------------------------------------
// CSA kernel: per spatial location p, 9-neighbor sigmoid correlation scores,
// softmax, weighted sum of out32 patches. One block (256 thr) per location.
// Output scattered into catT[q][512 + i] (second half of concat, K-major).
// ---------------------------------------------------------------------------
__global__ void k_csa(const float* __restrict__ sig, const float* __restrict__ out32,
                      float* __restrict__ catT) {
    const int p = blockIdx.x;
    const int h = p >> 5, w = p & 31;
    const int tid = threadIdx.x;
    __shared__ float part[9 * 256];
    __shared__ float attn[9];

    int qn[9];
#pragma unroll
    for (int d = 0; d < 9; ++d) {
        int dy = d / 3 - 1, dx = d % 3 - 1;
        int hh = h + dy, ww = w + dx;
        qn[d] = (hh >= 0 && hh < 32 && ww >= 0 && ww < 32) ? hh * 32 + ww : -1;
    }

    float partial[9] = {0, 0, 0, 0, 0, 0, 0, 0, 0};
    for (int c = tid; c < CC; c += 256) {
        float ctr = sig[c * HW + p];
#pragma unroll
        for (int d = 0; d < 9; ++d)
            if (qn[d] >= 0) partial[d] += ctr * sig[c * HW + qn[d]];
    }
#pragma unroll
    for (int d = 0; d < 9; ++d) part[d * 256 + tid] = partial[d];
    __syncthreads();
    for (int s = 128; s > 0; s >>= 1) {
        if (tid < s) {
#pragma unroll
            for (int d = 0; d < 9; ++d) part[d * 256 + tid] += part[d * 256 + tid + s];
        }
        __syncthreads();
    }
    if (tid == 0) {
        float sc[9], m = -1e30f;
#pragma unroll
        for (int d = 0; d < 9; ++d) {
            sc[d] = part[d * 256] * (1.0f / 512.0f);   // OOB stays 0 (zero-padded patch)
            m = fmaxf(m, sc[d]);
        }
        float se = 0.f;
#pragma unroll
        for (int d = 0; d < 9; ++d) { sc[d] = __expf(sc[d] - m); se += sc[d]; }
        float inv = 1.0f / se;
#pragma unroll
        for (int d = 0; d < 9; ++d) attn[d] = sc[d] * inv;
    }
    __syncthreads();
    for (int c = tid; c < CC; c += 256) {
        float a = 0.f;
#pragma unroll
        for (int d = 0; d < 9; ++d)
            if (qn[d] >= 0) a += attn[d] * out32[c * HW + qn[d]];
        int f = p * CC + c;                              // flat (p,c) index
        catT[((f & 1023) << 10) + 512 + (f >> 10)] = a;  // i = 512 + f/1024, q = f%1024
    }
}

// ---------------------------------------------------------------------------
// GEMM2 (WMMA f32): Y[o][q] = sum_i w_down[o][i] * catT[q][i]
// M=512(o), N=1024(q), K=1024(i). catT is K-major so B fragments are
// contiguous b64 loads, same as GEMM1. Wave = 64x64 tile (16 accumulators).
// grid (8, 2): m block = blockIdx.x; n block = blockIdx.y*8 + wave.
// ---------------------------------------------------------------------------
__global__ void __launch_bounds__(256, 1)
k_gemm2(const float* __restrict__ Wd, const float* __restrict__ catT,
        float* __restrict__ Y) {
    const int lane = threadIdx.x & 31;
    const int wave = threadIdx.x >> 5;
    const int half = lane >> 4;
    const int row = lane & 15;
    const int m0 = blockIdx.x * 64;
    const int n0 = (blockIdx.y * 8 + wave) * 64;

    const float* abase = Wd + (m0 + row) * HW + 2 * half;
    const float* bbase = catT + (n0 + row) * HW + 2 * half;

    v8f acc[4][4];
#pragma unroll
    for (int tm = 0; tm < 4; ++tm)
#pragma unroll
        for (int tn = 0; tn < 4; ++tn) acc[tm][tn] = (v8f){};

    for (int k = 0; k < HW; k += 4) {
        v2f a[4], b[4];
#pragma unroll
        for (int t = 0; t < 4; ++t) {
            a[t] = *(const v2f*)(abase + t * 16 * HW + k);
            b[t] = *(const v2f*)(bbase + t * 16 * HW + k);
        }
#pragma unroll
        for (int tm = 0; tm < 4; ++tm)
#pragma unroll
            for (int tn = 0; tn < 4; ++tn)
                acc[tm][tn] = WMMA_F32X4(a[tm], b[tn], acc[tm][tn]);
    }
#pragma unroll
    for (int tm = 0; tm < 4; ++tm) {
#pragma unroll
        for (int tn = 0; tn < 4; ++tn) {
#pragma unroll
            for (int r = 0; r < 8; ++r) {
                int o = m0 + tm * 16 + 8 * half + r;
                int q = n0 + tn * 16 + row;
                Y[o * HW + q] = acc[tm][tn][r];
            }
        }
    }
}

// ---------------------------------------------------------------------------
// InstanceNorm (affine=False, eps=1e-5, biased var) + LeakyReLU(0.2)
// ---------------------------------------------------------------------------
__global__ void k_inorm(const float* __restrict__ Y, float* __restrict__ out) {
    const int c = blockIdx.x;
    const int tid = threadIdx.x;
    __shared__ float rs[256], rq[256];
    float v[4];
    float s = 0.f, q = 0.f;
#pragma unroll
    for (int t = 0; t < 4; ++t) {
        float xv = Y[c * HW + tid + t * 256];
        v[t] = xv;
        s += xv;
        q += xv * xv;
    }
    rs[tid] = s;
    rq[tid] = q;
    __syncthreads();
    for (int st = 128; st > 0; st >>= 1) {
        if (tid < st) { rs[tid] += rs[tid + st]; rq[tid] += rq[tid + st]; }
        __syncthreads();
    }
    float mean = rs[0] * (1.0f / 1024.0f);
    float var = rq[0] * (1.0f / 1024.0f) - mean * mean;
    float inv = rsqrtf(var + 1e-5f);
#pragma unroll
    for (int t = 0; t < 4; ++t) {
        float val = (v[t] - mean) * inv;
        out[c * HW + tid + t * 256] = (val >= 0.f) ? val : 0.2f * val;
    }
}

// ---------------------------------------------------------------------------
extern "C" void kernel_launch(void* const* d_in, const int* in_sizes, int n_in,
                              void* d_out, int out_size, void* d_ws, size_t ws_size,
                              hipStream_t stream) {
    const float* x     = (const float*)d_in[0];  // (1,512,32,32)
    const float* w_se1 = (const float*)d_in[1];  // (32,512)
    const float* w_se2 = (const float*)d_in[2];  // (512,32)
    const float* w_dn  = (const float*)d_in[3];  // (512,1024,1,1) == (512,1024) flat

    float* ws = (float*)d_ws;
    float* G     = ws;                  // 1024*1024 gaussian weights
    float* sChan = G + 1024 * HW;       // 512
    float* s2    = sChan + 512;         // 512
    float* out32 = s2 + 512;            // 512*1024
    float* sig   = out32 + CC * HW;     // 512*1024
    float* catT  = sig + CC * HW;       // 1024*1024 K-major concat [q][i]
    float* Y     = G;                   // alias: G unused after GEMM1

    k_gauss<<<1024, 256, 0, stream>>>(G);
    k_mean <<<512, 256, 0, stream>>>(x, sChan);
    k_se   <<<1, 512, 0, stream>>>(sChan, w_se1, w_se2, s2);
    k_scale<<<2048, 256, 0, stream>>>(x, s2, out32, sig);
    k_gemm1<<<16, 256, 0, stream>>>(G, out32, catT);
    k_csa  <<<1024, 256, 0, stream>>>(sig, out32, catT);
    k_gemm2<<<dim3(8, 2), 256, 0, stream>>>(w_dn, catT, Y);
    k_inorm<<<512, 256, 0, stream>>>(Y, (float*)d_out);
}